// SinkhornDistance_70918499991956
// MI455X (gfx1250) — compile-verified
//
#include <hip/hip_runtime.h>
#include <math.h>

// Sinkhorn distance, N=4096, MI455X (gfx1250).
// M is L2-resident (64MB < 192MB L2); all sweeps after the first run at L2 BW.
// Column pass uses V_WMMA_F32_16X16X4_F32 (f32 matrix pipe, wave32 layout).

#define SK_N 4096
#define SK_NN ((size_t)SK_N * (size_t)SK_N)
#define SK_REG 0.5f
#define SK_EPS 1e-6f
#define SK_ITER 50

typedef float v2f __attribute__((ext_vector_type(2)));
typedef float v8f __attribute__((ext_vector_type(8)));

// ---------------------------------------------------------------- init ------
// M = exp(-C/REG); reset u=v=1, done=0, cost accumulator=0 (per-launch!).
__global__ void sk_init(const float* __restrict__ C, float* __restrict__ M,
                        float* __restrict__ u, float* __restrict__ v,
                        int* __restrict__ done, float* __restrict__ cost) {
    size_t i4 = (size_t)blockIdx.x * 256 + threadIdx.x;
    float4 c = ((const float4*)C)[i4];
    float4 m;
    const float k = -1.0f / SK_REG;
    m.x = __expf(k * c.x);
    m.y = __expf(k * c.y);
    m.z = __expf(k * c.z);
    m.w = __expf(k * c.w);
    ((float4*)M)[i4] = m;
    if (i4 < SK_N) { u[i4] = 1.0f; v[i4] = 1.0f; }
    if (i4 == 0)   { *done = 0; *cost = 0.0f; }
}

// ------------------------------------------------------------ row pass ------
// y = M @ v  (wave-per-row, float4 coalesced); u_new = src/y;
// w1=u^2, w2=u*u_new, w3=u_new^2; also zero the column-pass accumulators.
__global__ void sk_row(const float* __restrict__ M, const float* __restrict__ v,
                       const float* __restrict__ src, const float* __restrict__ u,
                       float* __restrict__ u_new,
                       float* __restrict__ w1, float* __restrict__ w2, float* __restrict__ w3,
                       float* __restrict__ z, float* __restrict__ b1,
                       float* __restrict__ b2, float* __restrict__ b3,
                       const int* __restrict__ done) {
    if (*done) return;
    __shared__ float sv[SK_N];
    int tid = threadIdx.x;
    for (int j = tid; j < SK_N; j += 256) sv[j] = v[j];
    int gid = blockIdx.x * 256 + tid;
    if (gid < SK_N) { z[gid] = 0.f; b1[gid] = 0.f; b2[gid] = 0.f; b3[gid] = 0.f; }
    __syncthreads();

    int lane = tid & 31;
    int row  = blockIdx.x * 8 + (tid >> 5);
    const float4* mrow = (const float4*)(M + (size_t)row * SK_N);
    const float4* sv4  = (const float4*)sv;
    float acc = 0.0f;
    for (int j = lane; j < SK_N / 4; j += 32) {
        float4 m = mrow[j];
        float4 w = sv4[j];
        acc += m.x * w.x + m.y * w.y + m.z * w.z + m.w * w.w;
    }
    for (int off = 16; off >= 1; off >>= 1) acc += __shfl_xor(acc, off, 32);
    if (lane == 0) {
        float un = src[row] / acc;
        u_new[row] = un;
        float uo = u[row];
        w1[row] = uo * uo;
        w2[row] = uo * un;
        w3[row] = un * un;
    }
}

// --------------------------------------------------------- column pass ------
// z_j  = sum_i M_ij * u'_i        (WMMA: B = 4x16 M tile, A = u' broadcast)
// bk_j = sum_i M_ij^2 * wk_i      (VALU on the same fragment values)
// Each wave owns 16 columns (j0) over a 256-row segment (k0); partials land
// via global atomics.  Grid: (32 tile-groups x 16 segments) x 256 threads.
__global__ void sk_col(const float* __restrict__ M,
                       const float* __restrict__ u_new,
                       const float* __restrict__ w1, const float* __restrict__ w2,
                       const float* __restrict__ w3,
                       float* __restrict__ z, float* __restrict__ b1,
                       float* __restrict__ b2, float* __restrict__ b3,
                       const int* __restrict__ done) {
    if (*done) return;
    __shared__ float su[256], s1[256], s2[256], s3[256];
    int tid = threadIdx.x;
    int k0  = blockIdx.y * 256;
    su[tid] = u_new[k0 + tid];
    s1[tid] = w1[k0 + tid];
    s2[tid] = w2[k0 + tid];
    s3[tid] = w3[k0 + tid];
    __syncthreads();

    int lane  = tid & 31;
    int j0    = (blockIdx.x * 8 + (tid >> 5)) * 16;
    int half2 = (lane >> 4) << 1;   // 0 for lanes 0-15 (K0/K1), 2 for 16-31 (K2/K3)
    int jl    = lane & 15;
    const float* base = M + (size_t)k0 * SK_N + (size_t)(j0 + jl);

    v8f acc = {0.f, 0.f, 0.f, 0.f, 0.f, 0.f, 0.f, 0.f};
    float pb1 = 0.f, pb2 = 0.f, pb3 = 0.f;
    for (int kk = 0; kk < 256; kk += 4) {
        int r0 = kk + half2;
        float m0 = base[(size_t)r0 * SK_N];          // B frag VGPR0 (row K0/K2)
        float m1 = base[(size_t)(r0 + 1) * SK_N];    // B frag VGPR1 (row K1/K3)
        v2f a; a.x = su[r0]; a.y = su[r0 + 1];       // A = u' broadcast per row
        v2f b; b.x = m0;     b.y = m1;
        acc = __builtin_amdgcn_wmma_f32_16x16x4_f32(
                false, a, false, b, (short)0, acc, false, false);
        float q0 = m0 * m0, q1 = m1 * m1;            // (M.^2) side reductions
        pb1 += q0 * s1[r0] + q1 * s1[r0 + 1];
        pb2 += q0 * s2[r0] + q1 * s2[r0 + 1];
        pb3 += q0 * s3[r0] + q1 * s3[r0 + 1];
    }
    pb1 += __shfl_xor(pb1, 16, 32);
    pb2 += __shfl_xor(pb2, 16, 32);
    pb3 += __shfl_xor(pb3, 16, 32);
    if (lane < 16) {                 // D row 0 = z partial for columns j0..j0+15
        atomicAdd(&z [j0 + jl], acc[0]);
        atomicAdd(&b1[j0 + jl], pb1);
        atomicAdd(&b2[j0 + jl], pb2);
        atomicAdd(&b3[j0 + jl], pb3);
    }
}

// -------------------------------------------------------------- update ------
// v' = tgt/z; err^2 = sum_j v^2 b1 - 2 v v' b2 + v'^2 b3; apply freeze logic.
__global__ void sk_update(const float* __restrict__ z, const float* __restrict__ b1,
                          const float* __restrict__ b2, const float* __restrict__ b3,
                          const float* __restrict__ tgt,
                          float* __restrict__ u, const float* __restrict__ u_new,
                          float* __restrict__ v, int* __restrict__ done) {
    if (*done) return;
    __shared__ float red[256];
    int tid = threadIdx.x;
    float e = 0.0f;
    for (int j = tid; j < SK_N; j += 256) {
        float vo = v[j];
        float vn = tgt[j] / z[j];
        e += vo * vo * b1[j] - 2.0f * vo * vn * b2[j] + vn * vn * b3[j];
        v[j] = vn;           // each j touched by exactly one thread
        u[j] = u_new[j];
    }
    red[tid] = e;
    __syncthreads();
    for (int s = 128; s > 0; s >>= 1) {
        if (tid < s) red[tid] += red[tid + s];
        __syncthreads();
    }
    if (tid == 0 && red[0] < SK_EPS * SK_EPS) *done = 1;
}

// --------------------------------------------------------------- final ------
// P = u .* M .* v -> out; cost = sum(P .* C) via block-reduce + one atomic.
__global__ void sk_final(const float* __restrict__ M, const float* __restrict__ C,
                         const float* __restrict__ u, const float* __restrict__ v,
                         float* __restrict__ P, float* __restrict__ cost) {
    __shared__ float red[256];
    size_t i4  = (size_t)blockIdx.x * 256 + threadIdx.x;
    size_t idx = i4 * 4;
    int row = (int)(idx >> 12);
    int col = (int)(idx & (SK_N - 1));
    float4 m  = ((const float4*)M)[i4];
    float4 c  = ((const float4*)C)[i4];
    float4 vv = ((const float4*)v)[col >> 2];
    float  ui = u[row];
    float4 p;
    p.x = ui * m.x * vv.x;
    p.y = ui * m.y * vv.y;
    p.z = ui * m.z * vv.z;
    p.w = ui * m.w * vv.w;
    ((float4*)P)[i4] = p;
    red[threadIdx.x] = p.x * c.x + p.y * c.y + p.z * c.z + p.w * c.w;
    __syncthreads();
    for (int s = 128; s > 0; s >>= 1) {
        if (threadIdx.x < s) red[threadIdx.x] += red[threadIdx.x + s];
        __syncthreads();
    }
    if (threadIdx.x == 0) atomicAdd(cost, red[0]);
}

// ---------------------------------------------------------------- host ------
extern "C" void kernel_launch(void* const* d_in, const int* in_sizes, int n_in,
                              void* d_out, int out_size, void* d_ws, size_t ws_size,
                              hipStream_t stream) {
    (void)in_sizes; (void)n_in; (void)out_size; (void)ws_size;
    const float* C   = (const float*)d_in[0];   // cost_matrix [N,N] f32
    const float* src = (const float*)d_in[1];   // source [N]
    const float* tgt = (const float*)d_in[2];   // target [N]
    float* P    = (float*)d_out;                // [N*N] then min_cost scalar
    float* cost = P + SK_NN;

    float* M     = (float*)d_ws;                // 64 MB, L2-resident
    float* u     = M + SK_NN;
    float* v     = u + SK_N;
    float* u_new = v + SK_N;
    float* w1 = u_new + SK_N; float* w2 = w1 + SK_N; float* w3 = w2 + SK_N;
    float* z  = w3 + SK_N;    float* b1 = z + SK_N;
    float* b2 = b1 + SK_N;    float* b3 = b2 + SK_N;
    int*   done = (int*)(b3 + SK_N);

    const int nb_full = (int)(SK_NN / 4 / 256);                  // 16384
    sk_init<<<nb_full, 256, 0, stream>>>(C, M, u, v, done, cost);
    for (int it = 0; it < SK_ITER; ++it) {
        sk_row<<<SK_N / 8, 256, 0, stream>>>(M, v, src, u, u_new,
                                             w1, w2, w3, z, b1, b2, b3, done);
        sk_col<<<dim3(SK_N / 16 / 8, SK_N / 256), 256, 0, stream>>>(
            M, u_new, w1, w2, w3, z, b1, b2, b3, done);
        sk_update<<<1, 256, 0, stream>>>(z, b1, b2, b3, tgt, u, u_new, v, done);
    }
    sk_final<<<nb_full, 256, 0, stream>>>(M, C, u, v, P, cost);
}